// LightGCN_82849919140548
// MI455X (gfx1250) — compile-verified
//
#include <hip/hip_runtime.h>
#include <hip/hip_bf16.h>
#include <stdint.h>

#define N_USERS 100000
#define N_ITEMS 50000
#define N_NODES (N_USERS + N_ITEMS)
#define D 64
#define NQ 4096
#define TILE 256
#define USE_ASYNC 1

typedef __attribute__((address_space(3))) uint8_t lds_u8_t;

static __device__ __forceinline__ uint32_t lds_off(const void* p) {
  // __shared__ pointers live in the LDS aperture; addrspacecast to AS(3)
  // + truncate yields the raw LDS byte offset for the async-to-LDS VDST.
  return (uint32_t)(uintptr_t)(lds_u8_t*)p;
}

// ---------------------------------------------------------------- k_init ----
// cur = concat(user_emb, item_emb); nxt = 0   (float4-vectorized)
__global__ __launch_bounds__(256) void k_init(const float* __restrict__ u,
                                              const float* __restrict__ it,
                                              float* __restrict__ cur,
                                              float* __restrict__ nxt) {
  const size_t total4 = (size_t)N_NODES * D / 4;
  const size_t ub4    = (size_t)N_USERS * D / 4;
  size_t i = (size_t)blockIdx.x * blockDim.x + threadIdx.x;
  const size_t stride = (size_t)gridDim.x * blockDim.x;
  const float4 z = {0.f, 0.f, 0.f, 0.f};
  for (; i < total4; i += stride) {
    float4 v = (i < ub4) ? ((const float4*)u)[i] : ((const float4*)it)[i - ub4];
    ((float4*)cur)[i] = v;
    ((float4*)nxt)[i] = z;
  }
}

// ---------------------------------------------------------------- k_zero ----
__global__ __launch_bounds__(256) void k_zero(float* __restrict__ p, size_t n4) {
  size_t i = (size_t)blockIdx.x * blockDim.x + threadIdx.x;
  const size_t stride = (size_t)gridDim.x * blockDim.x;
  const float4 z = {0.f, 0.f, 0.f, 0.f};
  for (; i < n4; i += stride) ((float4*)p)[i] = z;
}

// -------------------------------------------------------------- k_scatter ---
// nxt[src] += val * cur[dst] for every edge. One wave per edge; lane l owns
// the float2 at columns 2l,2l+1. Edge tuples are staged to LDS with gfx1250
// async-to-LDS loads, DOUBLE-BUFFERED: tile t+1 is in flight (ASYNCcnt)
// while tile t is being consumed by the gather/atomic loop.
struct TileBuf {
  int   src[TILE];
  int   dst[TILE];
  float val[TILE];
};

static __device__ __forceinline__ void stage_tile(TileBuf* b, int tid,
                                                  const int* __restrict__ esrc,
                                                  const int* __restrict__ edst,
                                                  const float* __restrict__ eval,
                                                  int64_t base, int n) {
  if (tid < n) {
#if USE_ASYNC
    uint32_t a0 = lds_off(&b->src[tid]);
    uint32_t a1 = lds_off(&b->dst[tid]);
    uint32_t a2 = lds_off(&b->val[tid]);
    asm volatile("global_load_async_to_lds_b32 %0, %1, off"
                 :: "v"(a0), "v"(esrc + base + tid) : "memory");
    asm volatile("global_load_async_to_lds_b32 %0, %1, off"
                 :: "v"(a1), "v"(edst + base + tid) : "memory");
    asm volatile("global_load_async_to_lds_b32 %0, %1, off"
                 :: "v"(a2), "v"(eval + base + tid) : "memory");
#else
    b->src[tid] = esrc[base + tid];
    b->dst[tid] = edst[base + tid];
    b->val[tid] = eval[base + tid];
#endif
  }
}

static __device__ __forceinline__ void async_wait_all() {
#if USE_ASYNC
  asm volatile("s_wait_asynccnt 0" ::: "memory");
#endif
}

__global__ __launch_bounds__(256) void k_scatter(const float* __restrict__ cur,
                                                 float* __restrict__ nxt,
                                                 const float* __restrict__ eval,
                                                 const int* __restrict__ esrc,
                                                 const int* __restrict__ edst,
                                                 int nE) {
  __shared__ TileBuf sb[2];

  const int tid  = threadIdx.x;   // 0..255
  const int lane = tid & 31;
  const int wv   = tid >> 5;      // wave 0..7

  const int64_t tileStride = (int64_t)gridDim.x * TILE;
  int64_t base = (int64_t)blockIdx.x * TILE;
  if (base >= nE) return;

  // Prologue: stage first tile and make it visible.
  stage_tile(&sb[0], tid, esrc, edst, eval,
             base, (int)min((int64_t)TILE, (int64_t)nE - base));
  async_wait_all();
  __syncthreads();

  int c = 0;
  for (; base < nE; base += tileStride, c ^= 1) {
    // Kick off the next tile's async copies before consuming this one.
    const int64_t nb = base + tileStride;
    if (nb < nE) {
      stage_tile(&sb[c ^ 1], tid, esrc, edst, eval,
                 nb, (int)min((int64_t)TILE, (int64_t)nE - nb));
    }

    const TileBuf* b = &sb[c];
    const int n = (int)min((int64_t)TILE, (int64_t)nE - base);

    if (n == TILE) {
      // Full tile: constant trip count (32 edges/wave) -> unroll for MLP.
#pragma unroll 4
      for (int e = wv; e < TILE; e += 8) {
        const int   src = __builtin_amdgcn_readfirstlane(b->src[e]);
        const int   dst = __builtin_amdgcn_readfirstlane(b->dst[e]);
        const float v   = b->val[e];
        const float2 r  = *((const float2*)(cur + (size_t)dst * D) + lane);
        float* o = nxt + (size_t)src * D + lane * 2;
        atomicAdd(o,     v * r.x);   // global_atomic_add_f32 (L2-resident)
        atomicAdd(o + 1, v * r.y);
      }
    } else {
      for (int e = wv; e < n; e += 8) {
        const int   src = __builtin_amdgcn_readfirstlane(b->src[e]);
        const int   dst = __builtin_amdgcn_readfirstlane(b->dst[e]);
        const float v   = b->val[e];
        const float2 r  = *((const float2*)(cur + (size_t)dst * D) + lane);
        float* o = nxt + (size_t)src * D + lane * 2;
        atomicAdd(o,     v * r.x);
        atomicAdd(o + 1, v * r.y);
      }
    }

    // Next tile must be fully landed (this wave) and everyone done reading
    // the buffer that iteration t+1 will overwrite.
    async_wait_all();
    __syncthreads();
  }
}

// --------------------------------------------------------------- k_gather ---
// out[t] (=|+=) 0.25 * emb[row(t)]  for 4096 user rows then 4096 item rows.
__global__ __launch_bounds__(256) void k_gather(const float* __restrict__ emb,
                                                const int* __restrict__ users,
                                                const int* __restrict__ items,
                                                float* __restrict__ out,
                                                int overwrite) {
  const int t = blockIdx.x * blockDim.x + threadIdx.x;  // 0..524287
  const int r = t >> 6;
  const int d = t & 63;
  const int node = (r < NQ) ? users[r] : (N_USERS + items[r - NQ]);
  const float v = 0.25f * emb[(size_t)node * D + d];
  if (overwrite) out[t] = v;
  else           out[t] += v;
}

// ----------------------------------------------------------------- launch ---
extern "C" void kernel_launch(void* const* d_in, const int* in_sizes, int n_in,
                              void* d_out, int out_size, void* d_ws, size_t ws_size,
                              hipStream_t stream) {
  const float* user_emb = (const float*)d_in[0];
  const float* item_emb = (const float*)d_in[1];
  const float* edge_val = (const float*)d_in[2];
  const int*   edge_src = (const int*)d_in[3];
  const int*   edge_dst = (const int*)d_in[4];
  const int*   users    = (const int*)d_in[5];
  const int*   items    = (const int*)d_in[6];
  float*       out      = (float*)d_out;
  const int    nE       = in_sizes[2];

  float* cur = (float*)d_ws;                       // 38.4 MB
  float* nxt = cur + (size_t)N_NODES * D;          // 38.4 MB (both L2-resident)

  k_init<<<2048, 256, 0, stream>>>(user_emb, item_emb, cur, nxt);
  k_gather<<<2048, 256, 0, stream>>>(cur, users, items, out, 1);  // layer-0 term

  for (int l = 0; l < 3; ++l) {
    k_scatter<<<4096, 256, 0, stream>>>(cur, nxt, edge_val, edge_src, edge_dst, nE);
    k_gather<<<2048, 256, 0, stream>>>(nxt, users, items, out, 0);
    float* t = cur; cur = nxt; nxt = t;            // propagate
    if (l < 2) k_zero<<<2048, 256, 0, stream>>>(nxt, (size_t)N_NODES * D / 4);
  }
}